// DBOW_20942260535953
// MI455X (gfx1250) — compile-verified
//
#include <hip/hip_runtime.h>

// DBOW scoring: out[b,k] = dot(D[doc_ids[b]], O[:, noise_ids[b,k]])
// Memory-bound (O fits in 192MB L2; ~60MB HBM traffic, ~0.11 GFLOP).
// One wave per batch element; dot products run on v_wmma_f32_16x16x4_f32
// (full f32 precision) with fragments built directly from the natural
// gather pattern of the strided O columns.

typedef __attribute__((ext_vector_type(2))) float v2f;
typedef __attribute__((ext_vector_type(8))) float v8f;

#define VEC_DIM        128
#define NUM_WORDS      100000
#define KCOLS          26
#define BATCH          16384
#define WAVES_PER_BLK  8

__global__ __launch_bounds__(256) void dbow_wmma_kernel(
    const int*   __restrict__ doc_ids,     // [BATCH]
    const int*   __restrict__ noise_ids,   // [BATCH, KCOLS]
    const float* __restrict__ D,           // [NUM_DOCS, VEC_DIM] row-major
    const float* __restrict__ O,           // [VEC_DIM, NUM_WORDS] row-major
    float*       __restrict__ out)         // [BATCH, KCOLS]
{
    const int lane = threadIdx.x & 31;
    const int wave = threadIdx.x >> 5;
    const int b    = blockIdx.x * WAVES_PER_BLK + wave;

    const int  col = lane & 15;       // N within a 16-wide tile
    const bool hi  = lane >= 16;      // high half-wave holds K=2,3 rows

    // Column words for tile0 (N=0..15) and tile1 (N=16..31, clamped to 25 so
    // the 6 pad lanes re-read word 25 -> loads broadcast-combine, ~free).
    const int w0 = noise_ids[b * KCOLS + col];
    const int c1 = 16 + col;
    const int w1 = noise_ids[b * KCOLS + (c1 < KCOLS ? c1 : (KCOLS - 1))];

    const float* __restrict__ drow = D + (size_t)doc_ids[b] * VEC_DIM;
    const float* __restrict__ col0 = O + w0;   // column base: elem v at v*NUM_WORDS
    const float* __restrict__ col1 = O + w1;

    v8f acc0 = {0.f, 0.f, 0.f, 0.f, 0.f, 0.f, 0.f, 0.f};
    v8f acc1 = {0.f, 0.f, 0.f, 0.f, 0.f, 0.f, 0.f, 0.f};

    // K loop: 4 v-values per WMMA, 32 steps for VEC_DIM=128.
    // Unroll so the gather loads of several steps are in flight together.
#pragma unroll 4
    for (int kk = 0; kk < VEC_DIM / 4; ++kk) {
        const int v0 = 4 * kk + (hi ? 2 : 0);

        // A fragment (16x4 f32): lane m = row m, VGPR0=K(v0), VGPR1=K(v0+1).
        // Only row M=0 (lanes 0 and 16) matters for the result; other rows
        // carry duplicated d-values which only feed D rows we never read.
        const v2f a = *(const v2f*)(drow + v0);   // 8B-aligned broadcast load

        // B fragments (4x16 f32): lane's word column at v0, v0+1.
        // These are the unavoidable 4B gathers (columns strided by 400KB),
        // served from L2 (O fits entirely in the 192MB L2).
        v2f b0, b1;
        b0.x = col0[(size_t)v0 * NUM_WORDS];
        b0.y = col0[(size_t)(v0 + 1) * NUM_WORDS];
        b1.x = col1[(size_t)v0 * NUM_WORDS];
        b1.y = col1[(size_t)(v0 + 1) * NUM_WORDS];

        // 8 args: (neg_a, A, neg_b, B, c_mod, C, reuse_a, reuse_b)
        acc0 = __builtin_amdgcn_wmma_f32_16x16x4_f32(
            false, a, false, b0, (short)0, acc0, false, false);
        acc1 = __builtin_amdgcn_wmma_f32_16x16x4_f32(
            false, a, false, b1, (short)0, acc1, false, false);
    }

    // D layout: VGPR0 of lanes 0..15 holds row M=0, N=lane.
    if (lane < 16) {
        out[b * KCOLS + lane] = acc0[0];
        if (16 + lane < KCOLS)
            out[b * KCOLS + 16 + lane] = acc1[0];
    }
}

extern "C" void kernel_launch(void* const* d_in, const int* in_sizes, int n_in,
                              void* d_out, int out_size, void* d_ws, size_t ws_size,
                              hipStream_t stream) {
    // setup_inputs order: context_ids(unused), doc_ids, target_noise_ids, D, O
    const int*   doc_ids   = (const int*)d_in[1];
    const int*   noise_ids = (const int*)d_in[2];
    const float* D         = (const float*)d_in[3];
    const float* O         = (const float*)d_in[4];
    float*       out       = (float*)d_out;

    dim3 grid(BATCH / WAVES_PER_BLK);   // 2048 blocks x 8 waves = 16384 waves
    dim3 block(256);
    dbow_wmma_kernel<<<grid, block, 0, stream>>>(doc_ids, noise_ids, D, O, out);
}